// DrugGraphEmbedding_61856118997222
// MI455X (gfx1250) — compile-verified
//
#include <hip/hip_runtime.h>
#include <hip/hip_bf16.h>

// Problem constants (match reference)
#define BB 32
#define PP 16
#define SS 256
#define DD 256
#define NG 128
#define NN (BB * NG)      // 4096 nodes
#define EE 65536          // edges (self-loops handled separately)
#define HH 256

typedef __attribute__((ext_vector_type(2))) float v2f;
typedef __attribute__((ext_vector_type(8))) float v8f;

// ---------------------------------------------------------------------------
// 1) Fused pathway-mean + gather: node_x[n][d] = mean_p emb[b][p][gid[n]][d]
//    grid = N blocks, 256 threads (d)
// ---------------------------------------------------------------------------
__global__ __launch_bounds__(256) void node_mean_kernel(
    const float* __restrict__ emb,   // [B,P,S,D]
    const int* __restrict__ gids,    // [B,NG] flat == [N]
    float* __restrict__ node_x)      // [N,D]
{
    const int n = blockIdx.x;
    const int d = threadIdx.x;
    const int b = n >> 7;            // n / NG
    const int g = gids[n];
    const float* base = emb + (((size_t)b * PP) * SS + g) * DD + d;
    float s = 0.f;
#pragma unroll
    for (int p = 0; p < PP; ++p)
        s += base[(size_t)p * SS * DD];
    node_x[n * DD + d] = s * (1.0f / PP);
}

// ---------------------------------------------------------------------------
// 2) Degree: init to 1 (self-loop), scatter-add over dst, then rsqrt in place
// ---------------------------------------------------------------------------
__global__ __launch_bounds__(256) void deg_init_kernel(float* __restrict__ deg)
{
    const int i = blockIdx.x * 256 + threadIdx.x;
    if (i < NN) deg[i] = 1.0f;
}

__global__ __launch_bounds__(256) void deg_scatter_kernel(
    const int* __restrict__ dst, float* __restrict__ deg)
{
    const int e = blockIdx.x * 256 + threadIdx.x;
    if (e < EE) atomicAdd(&deg[dst[e]], 1.0f);
}

__global__ __launch_bounds__(256) void deg_rsqrt_kernel(float* __restrict__ deg)
{
    const int i = blockIdx.x * 256 + threadIdx.x;
    if (i < NN) deg[i] = rsqrtf(deg[i]);
}

// ---------------------------------------------------------------------------
// 3) GEMM H = X @ W  with V_WMMA_F32_16X16X4_F32.
//    One wave per 16x16 output tile, K-loop in steps of 4.
//    X: [M,256] row-major, W: [256,256] row-major, H: [M,256].
//    grid: M/16 * 16 tiles -> (tiles*32)/256 blocks of 256 threads (8 waves).
// ---------------------------------------------------------------------------
__global__ __launch_bounds__(256) void gemm_wmma_f32_kernel(
    const float* __restrict__ X, const float* __restrict__ W,
    float* __restrict__ Hout, int M)
{
    const int wave = (blockIdx.x * 256 + threadIdx.x) >> 5;
    const int lane = threadIdx.x & 31;
    const int tilesN = HH / 16;              // 16
    const int tm = wave / tilesN;
    const int tn = wave % tilesN;

    const int m    = lane & 15;              // A row / D col index
    const int half = lane >> 4;              // 0: lanes 0-15, 1: lanes 16-31
    const int row  = tm * 16 + m;
    const int col  = tn * 16 + m;            // B/D column = lane % 16

    v8f acc = {};
    for (int k = 0; k < DD; k += 4) {
        const int kk = k + 2 * half;
        v2f a, b;
        // A 16x4 (MxK): lane m holds K = 2*half + {0,1}
        a.x = X[(size_t)row * DD + kk];
        a.y = X[(size_t)row * DD + kk + 1];
        // B 4x16 (KxN): lane n holds K = 2*half + {0,1}
        b.x = W[(size_t)kk * HH + col];
        b.y = W[(size_t)(kk + 1) * HH + col];
        acc = __builtin_amdgcn_wmma_f32_16x16x4_f32(
            /*neg_a=*/false, a, /*neg_b=*/false, b,
            /*c_mod=*/(short)0, acc, /*reuse_a=*/false, /*reuse_b=*/false);
    }
#pragma unroll
    for (int r = 0; r < 8; ++r) {
        const int orow = tm * 16 + r + 8 * half;
        Hout[(size_t)orow * HH + col] = acc[r];
    }
}

// ---------------------------------------------------------------------------
// 4) Aggregation init with self-loop message: out[n][d] = dis[n]^2 * h[n][d]
// ---------------------------------------------------------------------------
__global__ __launch_bounds__(256) void agg_init_kernel(
    const float* __restrict__ h, const float* __restrict__ dis,
    float* __restrict__ out)
{
    const int n = blockIdx.x;
    const int d = threadIdx.x;
    const float w = dis[n] * dis[n];
    out[n * HH + d] = w * h[n * HH + d];
}

// ---------------------------------------------------------------------------
// 5) Edge scatter: out[dst][d] += dis[src]*dis[dst]*h[src][d]
//    grid = E blocks, 256 threads; targets stay resident in L2 (4 MB).
// ---------------------------------------------------------------------------
__global__ __launch_bounds__(256) void edge_scatter_kernel(
    const int* __restrict__ src, const int* __restrict__ dst,
    const float* __restrict__ dis, const float* __restrict__ h,
    float* __restrict__ out)
{
    const int e = blockIdx.x;
    const int d = threadIdx.x;
    const int u = src[e];
    const int v = dst[e];
    const float w = dis[u] * dis[v];
    atomicAdd(&out[(size_t)v * HH + d], w * h[(size_t)u * HH + d]);
}

// ---------------------------------------------------------------------------
// 6) Bias (+ optional ReLU), in place
// ---------------------------------------------------------------------------
__global__ __launch_bounds__(256) void bias_act_kernel(
    const float* __restrict__ bias, float* __restrict__ x, int do_relu)
{
    const int n = blockIdx.x;
    const int d = threadIdx.x;
    float v = x[n * HH + d] + bias[d];
    if (do_relu) v = fmaxf(v, 0.0f);
    x[n * HH + d] = v;
}

// ---------------------------------------------------------------------------
// 7) Global mean pool per graph: out[b][d] = mean_g x[b*NG+g][d]
// ---------------------------------------------------------------------------
__global__ __launch_bounds__(256) void pool_kernel(
    const float* __restrict__ x, float* __restrict__ out)
{
    const int b = blockIdx.x;
    const int d = threadIdx.x;
    float s = 0.f;
#pragma unroll 4
    for (int g = 0; g < NG; ++g)
        s += x[(size_t)(b * NG + g) * HH + d];
    out[b * HH + d] = s * (1.0f / NG);
}

// ---------------------------------------------------------------------------
extern "C" void kernel_launch(void* const* d_in, const int* in_sizes, int n_in,
                              void* d_out, int out_size, void* d_ws, size_t ws_size,
                              hipStream_t stream)
{
    const float* emb  = (const float*)d_in[0];   // [B,P,S,D]
    const int*   gids = (const int*)d_in[1];     // [B,NG]
    const int*   eidx = (const int*)d_in[2];     // [2,E]
    const float* W1   = (const float*)d_in[3];
    const float* b1   = (const float*)d_in[4];
    const float* W2   = (const float*)d_in[5];
    const float* b2   = (const float*)d_in[6];
    float* out = (float*)d_out;                  // [B,H]

    const int* src = eidx;
    const int* dst = eidx + EE;

    // Workspace layout (fp32): node_x | h | agg | dis
    float* node_x = (float*)d_ws;                // N*D   = 4 MB
    float* hbuf   = node_x + (size_t)NN * DD;    // N*H   = 4 MB
    float* aggbuf = hbuf   + (size_t)NN * HH;    // N*H   = 4 MB
    float* dis    = aggbuf + (size_t)NN * HH;    // N     = 16 KB
    float* out2   = node_x;                      // reuse (node_x dead after GEMM1)

    const int gemm_blocks = ((NN / 16) * (HH / 16) * 32) / 256;  // 512

    // 1) features
    node_mean_kernel<<<NN, 256, 0, stream>>>(emb, gids, node_x);

    // 2) degrees -> dis = rsqrt(deg)
    deg_init_kernel<<<NN / 256, 256, 0, stream>>>(dis);
    deg_scatter_kernel<<<EE / 256, 256, 0, stream>>>(dst, dis);
    deg_rsqrt_kernel<<<NN / 256, 256, 0, stream>>>(dis);

    // 3) layer 1: h = node_x @ W1 ; aggregate ; + b1 ; ReLU
    gemm_wmma_f32_kernel<<<gemm_blocks, 256, 0, stream>>>(node_x, W1, hbuf, NN);
    agg_init_kernel<<<NN, 256, 0, stream>>>(hbuf, dis, aggbuf);
    edge_scatter_kernel<<<EE, 256, 0, stream>>>(src, dst, dis, hbuf, aggbuf);
    bias_act_kernel<<<NN, 256, 0, stream>>>(b1, aggbuf, 1);

    // 4) layer 2: h = x1 @ W2 ; aggregate ; + b2
    gemm_wmma_f32_kernel<<<gemm_blocks, 256, 0, stream>>>(aggbuf, W2, hbuf, NN);
    agg_init_kernel<<<NN, 256, 0, stream>>>(hbuf, dis, out2);
    edge_scatter_kernel<<<EE, 256, 0, stream>>>(src, dst, dis, hbuf, out2);
    bias_act_kernel<<<NN, 256, 0, stream>>>(b2, out2, 0);

    // 5) mean pool -> [B,H]
    pool_kernel<<<BB, 256, 0, stream>>>(out2, out);
}